// emdDist_28217935135201
// MI455X (gfx1250) — compile-verified
//
#include <hip/hip_runtime.h>
#include <math.h>

// EMD approxmatch (Fan et al.) for MI455X / gfx1250.
// 10 annealed soft-assignment iterations; match matrix never materialized.
// d^2 (or l2*d^2) tiles produced entirely by V_WMMA_F32_16X16X4_F32:
//   A = -2*(l2)*x1 rows (K=3 padded to 4), B = x2 cols, C = norms -> D = exp2 arg.

typedef __attribute__((ext_vector_type(2))) float v2f;
typedef __attribute__((ext_vector_type(8))) float v8f;

#define EMD_EPS 1e-9f
#define BLOCK_THREADS 256
#define WAVES_PER_BLOCK 8
#define ROWS_PER_BLOCK 128   // 8 waves * 16 rows

static __device__ __forceinline__ float fast_exp2(float x) {
#if __has_builtin(__builtin_amdgcn_exp2f)
  return __builtin_amdgcn_exp2f(x);
#else
  return exp2f(x);
#endif
}

// D = A(16x4 f32) * B(4x16 f32) + C(16x16 f32). wave32 only.
static __device__ __forceinline__ v8f wmma_k4(v2f a, v2f b, v8f c) {
  return __builtin_amdgcn_wmma_f32_16x16x4_f32(
      /*neg_a=*/false, a, /*neg_b=*/false, b,
      /*c_mod=*/(short)0, c, /*reuse_a=*/false, /*reuse_b=*/false);
}

// ---------------------------------------------------------------------------
// init: remainL = factorl, remainR = factorr, cost = 0
// ---------------------------------------------------------------------------
__global__ void emd_init(float* remainL, float* remainR, float* cost,
                         int B, int n, int m, float factorl, float factorr) {
  int tid = blockIdx.x * blockDim.x + threadIdx.x;
  int total = B * n + B * m;
  if (tid < B * n) remainL[tid] = factorl;
  else if (tid < total) remainR[tid - B * n] = factorr;
  if (tid < B) cost[tid] = 0.0f;
}

// ---------------------------------------------------------------------------
// pass A (row-oriented): s[b,k] = sum_l exp(level*d2[k,l]) * remainR[b,l]
// l2 scale folded into A and the staged norms; WMMA C carries the norms.
// ---------------------------------------------------------------------------
__global__ void __launch_bounds__(BLOCK_THREADS)
emd_passA(const float* __restrict__ x1, const float* __restrict__ x2,
          const float* __restrict__ remainR, float* __restrict__ s_out,
          float l2, int n, int m) {
  extern __shared__ float lds[];
  float* ldsX2 = lds;            // 4*m  (x,y,z,0 padded)
  float* ldsNR = lds + 4 * m;    // 2*m  interleaved (l2*|x2|^2, remainR)
  float* ldsN1 = lds + 6 * m;    // ROWS_PER_BLOCK (l2*|x1|^2)

  const int chunks = n / ROWS_PER_BLOCK;
  const int b = blockIdx.x / chunks;
  const int rchunk = blockIdx.x % chunks;
  const int tid = threadIdx.x;
  const int lane = tid & 31;
  const int wave = tid >> 5;
  const int hi = lane >> 4;   // half-wave: 0 or 1
  const int ln = lane & 15;
  const int k0 = hi * 2;      // K base for A/B VGPR layout

  const float* x1b = x1 + (size_t)b * n * 3;
  const float* x2b = x2 + (size_t)b * m * 3;

  for (int i = tid; i < m; i += BLOCK_THREADS) {
    float xx = x2b[3 * i], yy = x2b[3 * i + 1], zz = x2b[3 * i + 2];
    ldsX2[4 * i] = xx; ldsX2[4 * i + 1] = yy;
    ldsX2[4 * i + 2] = zz; ldsX2[4 * i + 3] = 0.0f;
    ldsNR[2 * i] = l2 * (xx * xx + yy * yy + zz * zz);
    ldsNR[2 * i + 1] = remainR[(size_t)b * m + i];
  }
  for (int i = tid; i < ROWS_PER_BLOCK; i += BLOCK_THREADS) {
    int r = rchunk * ROWS_PER_BLOCK + i;
    float xx = x1b[3 * r], yy = x1b[3 * r + 1], zz = x1b[3 * r + 2];
    ldsN1[i] = l2 * (xx * xx + yy * yy + zz * zz);
  }
  __syncthreads();

  const int row0 = rchunk * ROWS_PER_BLOCK + wave * 16;
  const int myrow = row0 + ln;
  const float c2 = -2.0f * l2;
  v2f A;  // A = -2*l2 * x1 rows (16x4, K=3 padded with 0)
  A.x = c2 * x1b[3 * myrow + k0];
  A.y = (k0 + 1 < 3) ? (c2 * x1b[3 * myrow + k0 + 1]) : 0.0f;

  float n1v[8];
#pragma unroll
  for (int v = 0; v < 8; ++v) n1v[v] = ldsN1[wave * 16 + v + 8 * hi];

  v8f acc = {};
  for (int c0 = 0; c0 < m; c0 += 16) {
    int col = c0 + ln;
    v2f Bm = *(const v2f*)(ldsX2 + 4 * col + k0);           // one ds_load_b64
    float2 nr = ((const float2*)ldsNR)[col];                // (l2*n2, remainR)
    v8f c;
#pragma unroll
    for (int v = 0; v < 8; ++v) c[v] = n1v[v] + nr.x;
    v8f d = wmma_k4(A, Bm, c);  // d = l2 * d2  (full exp2 argument)
#pragma unroll
    for (int v = 0; v < 8; ++v) acc[v] += fast_exp2(d[v]) * nr.y;
  }
  // row sums: reduce across each 16-lane half
#pragma unroll
  for (int v = 0; v < 8; ++v) {
    float a = acc[v];
    a += __shfl_xor(a, 1);
    a += __shfl_xor(a, 2);
    a += __shfl_xor(a, 4);
    a += __shfl_xor(a, 8);
    acc[v] = a;
  }
  if (ln == 0) {
#pragma unroll
    for (int v = 0; v < 8; ++v)
      s_out[(size_t)b * n + row0 + v + 8 * hi] = acc[v];
  }
}

// ---------------------------------------------------------------------------
// pass B (column-oriented): colacc[l] = sum_k exp(level*d2)*remainL[k]/(s[k]+eps)
// then ratio/remainR update in closed form; writes ratioEff = ratio*remainR_old.
// ---------------------------------------------------------------------------
__global__ void __launch_bounds__(BLOCK_THREADS)
emd_passB(const float* __restrict__ x1, const float* __restrict__ x2,
          const float* __restrict__ remainL, const float* __restrict__ s_in,
          float* __restrict__ remainR, float* __restrict__ ratioEff,
          float l2, int n, int m) {
  extern __shared__ float lds[];
  float* ldsX1 = lds;            // 4*n  (x,y,z,0 padded)
  float* ldsNW = lds + 4 * n;    // 2*n  interleaved (l2*|x1|^2, remainL/(s+eps))

  const int chunks = m / ROWS_PER_BLOCK;
  const int b = blockIdx.x / chunks;
  const int cchunk = blockIdx.x % chunks;
  const int tid = threadIdx.x;
  const int lane = tid & 31;
  const int wave = tid >> 5;
  const int hi = lane >> 4;
  const int ln = lane & 15;
  const int k0 = hi * 2;

  const float* x1b = x1 + (size_t)b * n * 3;
  const float* x2b = x2 + (size_t)b * m * 3;

  for (int i = tid; i < n; i += BLOCK_THREADS) {
    float xx = x1b[3 * i], yy = x1b[3 * i + 1], zz = x1b[3 * i + 2];
    ldsX1[4 * i] = xx; ldsX1[4 * i + 1] = yy;
    ldsX1[4 * i + 2] = zz; ldsX1[4 * i + 3] = 0.0f;
    ldsNW[2 * i] = l2 * (xx * xx + yy * yy + zz * zz);
    ldsNW[2 * i + 1] =
        remainL[(size_t)b * n + i] / (s_in[(size_t)b * n + i] + EMD_EPS);
  }
  __syncthreads();

  const int col0 = cchunk * ROWS_PER_BLOCK + wave * 16;
  const int mycol = col0 + ln;
  v2f Bm;  // fixed B = this wave's 16 x2 columns
  Bm.x = x2b[3 * mycol + k0];
  Bm.y = (k0 + 1 < 3) ? x2b[3 * mycol + k0 + 1] : 0.0f;
  float cx = x2b[3 * mycol], cy = x2b[3 * mycol + 1], cz = x2b[3 * mycol + 2];
  float n2s = l2 * (cx * cx + cy * cy + cz * cz);
  const float c2 = -2.0f * l2;

  float colacc = 0.0f;
  for (int r0 = 0; r0 < n; r0 += 16) {
    int row = r0 + ln;
    v2f Am = *(const v2f*)(ldsX1 + 4 * row + k0);
    Am.x *= c2; Am.y *= c2;
    v8f c;
    float wv[8];
#pragma unroll
    for (int v = 0; v < 8; ++v) {
      float2 nw = ((const float2*)ldsNW)[r0 + v + 8 * hi];
      c[v] = nw.x + n2s;
      wv[v] = nw.y;
    }
    v8f d = wmma_k4(Am, Bm, c);  // l2 * d2
#pragma unroll
    for (int v = 0; v < 8; ++v) colacc += fast_exp2(d[v]) * wv[v];
  }
  colacc += __shfl_xor(colacc, 16);  // combine half-waves (same column set)
  if (hi == 0) {
    size_t gi = (size_t)b * m + mycol;
    float rrv = remainR[gi];
    float sumr = rrv * colacc + EMD_EPS;        // sum_k w + eps
    float ratio = fminf(rrv / sumr, 1.0f);
    ratioEff[gi] = ratio * rrv;
    float newrr = rrv - ratio * rrv * colacc;   // remainR - sum_k delta
    remainR[gi] = fmaxf(newrr, 0.0f);
  }
}

// ---------------------------------------------------------------------------
// pass C (row-oriented): delta = exp * (remainL/(s+eps)) * ratioEff[l]
// needs raw d2 for sqrt -> norms folded into WMMA C, l2 applied after.
// ---------------------------------------------------------------------------
__global__ void __launch_bounds__(BLOCK_THREADS)
emd_passC(const float* __restrict__ x1, const float* __restrict__ x2,
          const float* __restrict__ s_in, const float* __restrict__ ratioEff,
          float* __restrict__ remainL, float* __restrict__ costPartial,
          float l2, int n, int m) {
  extern __shared__ float lds[];
  float* ldsX2 = lds;            // 4*m  (x,y,z,0 padded)
  float* ldsNR = lds + 4 * m;    // 2*m  interleaved (|x2|^2, ratioEff)
  float* ldsN1 = lds + 6 * m;    // ROWS_PER_BLOCK (|x1|^2)
  float* ldsPart = lds + 6 * m + ROWS_PER_BLOCK;  // 16 partial cost values

  const int chunks = n / ROWS_PER_BLOCK;
  const int b = blockIdx.x / chunks;
  const int rchunk = blockIdx.x % chunks;
  const int tid = threadIdx.x;
  const int lane = tid & 31;
  const int wave = tid >> 5;
  const int hi = lane >> 4;
  const int ln = lane & 15;
  const int k0 = hi * 2;

  const float* x1b = x1 + (size_t)b * n * 3;
  const float* x2b = x2 + (size_t)b * m * 3;

  for (int i = tid; i < m; i += BLOCK_THREADS) {
    float xx = x2b[3 * i], yy = x2b[3 * i + 1], zz = x2b[3 * i + 2];
    ldsX2[4 * i] = xx; ldsX2[4 * i + 1] = yy;
    ldsX2[4 * i + 2] = zz; ldsX2[4 * i + 3] = 0.0f;
    ldsNR[2 * i] = xx * xx + yy * yy + zz * zz;
    ldsNR[2 * i + 1] = ratioEff[(size_t)b * m + i];
  }
  for (int i = tid; i < ROWS_PER_BLOCK; i += BLOCK_THREADS) {
    int r = rchunk * ROWS_PER_BLOCK + i;
    float xx = x1b[3 * r], yy = x1b[3 * r + 1], zz = x1b[3 * r + 2];
    ldsN1[i] = xx * xx + yy * yy + zz * zz;
  }
  __syncthreads();

  const int row0 = rchunk * ROWS_PER_BLOCK + wave * 16;
  const int myrow = row0 + ln;
  v2f A;  // A = -2 * x1 rows (unscaled; need raw d2)
  A.x = -2.0f * x1b[3 * myrow + k0];
  A.y = (k0 + 1 < 3) ? (-2.0f * x1b[3 * myrow + k0 + 1]) : 0.0f;

  float n1v[8];
#pragma unroll
  for (int v = 0; v < 8; ++v) n1v[v] = ldsN1[wave * 16 + v + 8 * hi];

  v8f sum1 = {};  // sum exp*ratioEff          -> rowdelta
  v8f sum2 = {};  // sum exp*ratioEff*dist     -> cost
  for (int c0 = 0; c0 < m; c0 += 16) {
    int col = c0 + ln;
    v2f Bm = *(const v2f*)(ldsX2 + 4 * col + k0);
    float2 nr = ((const float2*)ldsNR)[col];  // (n2, ratioEff)
    v8f c;
#pragma unroll
    for (int v = 0; v < 8; ++v) c[v] = n1v[v] + nr.x;
    v8f d = wmma_k4(A, Bm, c);  // raw d2
#pragma unroll
    for (int v = 0; v < 8; ++v) {
      float d2 = d[v];
      float e = fast_exp2(l2 * d2) * nr.y;
      sum1[v] += e;
      sum2[v] += e * sqrtf(fmaxf(d2, 1e-12f));
    }
  }
#pragma unroll
  for (int v = 0; v < 8; ++v) {
    float a = sum1[v], c = sum2[v];
    a += __shfl_xor(a, 1); c += __shfl_xor(c, 1);
    a += __shfl_xor(a, 2); c += __shfl_xor(c, 2);
    a += __shfl_xor(a, 4); c += __shfl_xor(c, 4);
    a += __shfl_xor(a, 8); c += __shfl_xor(c, 8);
    sum1[v] = a; sum2[v] = c;
  }
  if (ln == 0) {
    float csum = 0.0f;
#pragma unroll
    for (int v = 0; v < 8; ++v) {
      int row = row0 + v + 8 * hi;
      size_t gi = (size_t)b * n + row;
      float rl = remainL[gi];
      float invs = rl / (s_in[gi] + EMD_EPS);
      remainL[gi] = fmaxf(rl - invs * sum1[v], 0.0f);
      csum += invs * sum2[v];
    }
    ldsPart[wave * 2 + hi] = csum;
  }
  __syncthreads();
  if (tid == 0) {
    float t = 0.0f;
#pragma unroll
    for (int i = 0; i < 2 * WAVES_PER_BLOCK; ++i) t += ldsPart[i];
    costPartial[blockIdx.x] = t;
  }
}

// cost[b] += sum over chunk partials (deterministic order, no atomics)
__global__ void emd_finalize(const float* __restrict__ costPartial,
                             float* __restrict__ cost, int B, int chunks) {
  int b = blockIdx.x * blockDim.x + threadIdx.x;
  if (b < B) {
    float t = 0.0f;
    for (int i = 0; i < chunks; ++i) t += costPartial[b * chunks + i];
    cost[b] += t;
  }
}

// ---------------------------------------------------------------------------
extern "C" void kernel_launch(void* const* d_in, const int* in_sizes, int n_in,
                              void* d_out, int out_size, void* d_ws, size_t ws_size,
                              hipStream_t stream) {
  (void)n_in; (void)ws_size;
  const float* x1 = (const float*)d_in[0];
  const float* x2 = (const float*)d_in[1];
  float* cost = (float*)d_out;

  const int B = out_size;
  const int n = in_sizes[0] / (3 * B);
  const int m = in_sizes[1] / (3 * B);
  const int chunksA = n / ROWS_PER_BLOCK;
  const int chunksB = m / ROWS_PER_BLOCK;

  float* ws = (float*)d_ws;
  float* remainL  = ws;                                  // B*n
  float* remainR  = remainL + (size_t)B * n;             // B*m
  float* s        = remainR + (size_t)B * m;             // B*n
  float* ratioEff = s + (size_t)B * n;                   // B*m
  float* costPart = ratioEff + (size_t)B * m;            // B*chunksA

  const float factorl = (float)((n > m) ? n : m) / (float)n;
  const float factorr = (float)((n > m) ? n : m) / (float)m;

  {
    int total = B * n + B * m;
    int blocks = (total + 255) / 256;
    emd_init<<<blocks, 256, 0, stream>>>(remainL, remainR, cost, B, n, m,
                                         factorl, factorr);
  }

  const size_t ldsA = (size_t)(6 * m + ROWS_PER_BLOCK + 16) * sizeof(float);
  const size_t ldsB = (size_t)(6 * n) * sizeof(float);
  dim3 gridA(B * chunksA), gridB(B * chunksB);

  for (int j = 7; j >= -2; --j) {
    float level = (j == -2) ? 0.0f : -ldexpf(1.0f, 2 * j);  // -(4^j)
    float l2 = level * 1.4426950408889634f;                 // to exp2 domain
    emd_passA<<<gridA, BLOCK_THREADS, ldsA, stream>>>(x1, x2, remainR, s, l2, n, m);
    emd_passB<<<gridB, BLOCK_THREADS, ldsB, stream>>>(x1, x2, remainL, s,
                                                      remainR, ratioEff, l2, n, m);
    emd_passC<<<gridA, BLOCK_THREADS, ldsA, stream>>>(x1, x2, s, ratioEff,
                                                      remainL, costPart, l2, n, m);
    emd_finalize<<<1, 256, 0, stream>>>(costPart, cost, B, chunksA);
  }
}